// GausRast_47132971107148
// MI455X (gfx1250) — compile-verified
//
#include <hip/hip_runtime.h>
#include <math.h>

typedef __attribute__((ext_vector_type(2))) float v2f;
typedef __attribute__((ext_vector_type(8))) float v8f;

#define NPTS 2048
#define H_IMG 128
#define W_IMG 128
#define NPIX (H_IMG * W_IMG)
#define CHUNK 16
#define PSTRIDE 16   // floats per gaussian record: c0..c7, op, r,g,b, pad[4]
#define NCHUNK (NPTS / CHUNK)
#define CHUNK_BYTES (CHUNK * PSTRIDE * 4)   // 1024 B per chunk

// ---------------------------------------------------------------------------
// Kernel 1: per-gaussian preprocessing.
// Produces polynomial coefficients of power(x,y) = sum_k coeff[k]*feat[k],
// feat = [x^2, y^2, x*y, x, y, 1, 0, 0], plus effective opacity (valid folded
// in: op=0 when invalid -> identical blend result) and color.
// Writes radii and depth (original order) directly into d_out.
// ---------------------------------------------------------------------------
__global__ __launch_bounds__(256) void gs_preprocess(
    const float* __restrict__ pts, const float* __restrict__ scales,
    const float* __restrict__ colors, const float* __restrict__ opac,
    const float* __restrict__ rots, const float* __restrict__ view,
    const float* __restrict__ proj,
    float* __restrict__ params, float* __restrict__ out_radii,
    float* __restrict__ out_depth)
{
    int i = blockIdx.x * blockDim.x + threadIdx.x;
    if (i >= NPTS) return;

    float p0 = pts[3*i+0], p1 = pts[3*i+1], p2 = pts[3*i+2];
    float s0 = expf(scales[3*i+0]), s1 = expf(scales[3*i+1]), s2 = expf(scales[3*i+2]);
    float cr = 1.0f / (1.0f + expf(-colors[3*i+0]));
    float cg = 1.0f / (1.0f + expf(-colors[3*i+1]));
    float cb = 1.0f / (1.0f + expf(-colors[3*i+2]));
    float op = 1.0f / (1.0f + expf(-opac[i]));

    // quaternion -> rotation
    float q0 = rots[4*i+0], q1 = rots[4*i+1], q2 = rots[4*i+2], q3 = rots[4*i+3];
    float qn = rsqrtf(q0*q0 + q1*q1 + q2*q2 + q3*q3);
    float r = q0*qn, x = q1*qn, y = q2*qn, z = q3*qn;
    float R00 = 1.f - 2.f*(y*y + z*z), R01 = 2.f*(x*y - r*z), R02 = 2.f*(x*z + r*y);
    float R10 = 2.f*(x*y + r*z), R11 = 1.f - 2.f*(x*x + z*z), R12 = 2.f*(y*z - r*x);
    float R20 = 2.f*(x*z - r*y), R21 = 2.f*(y*z + r*x), R22 = 1.f - 2.f*(x*x + y*y);

    // M = R * diag(s); Sigma = M M^T
    float M00 = R00*s0, M01 = R01*s1, M02 = R02*s2;
    float M10 = R10*s0, M11 = R11*s1, M12 = R12*s2;
    float M20 = R20*s0, M21 = R21*s1, M22 = R22*s2;
    float S00 = M00*M00 + M01*M01 + M02*M02;
    float S01 = M00*M10 + M01*M11 + M02*M12;
    float S02 = M00*M20 + M01*M21 + M02*M22;
    float S11 = M10*M10 + M11*M11 + M12*M12;
    float S12 = M10*M20 + M11*M21 + M12*M22;
    float S22 = M20*M20 + M21*M21 + M22*M22;

    // camera-space position
    float t0 = view[0]*p0 + view[1]*p1 + view[2]*p2  + view[3];
    float t1 = view[4]*p0 + view[5]*p1 + view[6]*p2  + view[7];
    float tz = view[8]*p0 + view[9]*p1 + view[10]*p2 + view[11];

    const float fx = (float)W_IMG * 0.5f;   // W/(2*TANX)
    const float fy = (float)H_IMG * 0.5f;
    const float limx = 1.3f, limy = 1.3f;
    float txz = fminf(fmaxf(t0 / tz, -limx), limx) * tz;
    float tyz = fminf(fmaxf(t1 / tz, -limy), limy) * tz;
    float itz = 1.0f / tz;
    float J00 = fx * itz, J02 = -fx * txz * itz * itz;
    float J11 = fy * itz, J12 = -fy * tyz * itz * itz;

    // Tm = J @ view[:3,:3] (2x3)
    float T00 = J00*view[0] + J02*view[8];
    float T01 = J00*view[1] + J02*view[9];
    float T02 = J00*view[2] + J02*view[10];
    float T10 = J11*view[4] + J12*view[8];
    float T11 = J11*view[5] + J12*view[9];
    float T12 = J11*view[6] + J12*view[10];

    // cov = Tm Sigma Tm^T (2x2 symmetric)
    float u0x = S00*T00 + S01*T01 + S02*T02;
    float u0y = S01*T00 + S11*T01 + S12*T02;
    float u0z = S02*T00 + S12*T01 + S22*T02;
    float u1x = S00*T10 + S01*T11 + S02*T12;
    float u1y = S01*T10 + S11*T11 + S12*T12;
    float u1z = S02*T10 + S12*T11 + S22*T12;
    float cov00 = T00*u0x + T01*u0y + T02*u0z;
    float cov01 = T00*u1x + T01*u1y + T02*u1z;
    float cov11 = T10*u1x + T11*u1y + T12*u1z;

    float a = cov00 + 0.3f;
    float b = cov01;
    float c = cov11 + 0.3f;
    float det = a*c - b*b;
    bool valid = (tz > 0.2f) && (det > 0.0f);
    float det_s = (det > 0.0f) ? det : 1.0f;
    float cA = c / det_s, cB = -b / det_s, cC = a / det_s;
    float mid = 0.5f * (a + c);
    float lam = mid + sqrtf(fmaxf(mid*mid - det, 0.1f));
    float radii = valid ? ceilf(3.0f * sqrtf(lam)) : 0.0f;

    // full = proj @ view; ph = full[:, :3] @ p + full[:,3]
    float ph[4];
#pragma unroll
    for (int rr2 = 0; rr2 < 4; ++rr2) {
        float f0 = proj[rr2*4+0]*view[0] + proj[rr2*4+1]*view[4] + proj[rr2*4+2]*view[8]  + proj[rr2*4+3]*view[12];
        float f1 = proj[rr2*4+0]*view[1] + proj[rr2*4+1]*view[5] + proj[rr2*4+2]*view[9]  + proj[rr2*4+3]*view[13];
        float f2 = proj[rr2*4+0]*view[2] + proj[rr2*4+1]*view[6] + proj[rr2*4+2]*view[10] + proj[rr2*4+3]*view[14];
        float f3 = proj[rr2*4+0]*view[3] + proj[rr2*4+1]*view[7] + proj[rr2*4+2]*view[11] + proj[rr2*4+3]*view[15];
        ph[rr2] = f0*p0 + f1*p1 + f2*p2 + f3;
    }
    float pw = 1.0f / (ph[3] + 1e-7f);
    float px = ((ph[0]*pw + 1.0f) * (float)W_IMG - 1.0f) * 0.5f;
    float py = ((ph[1]*pw + 1.0f) * (float)H_IMG - 1.0f) * 0.5f;

    // polynomial expansion of -0.5*(cA dx^2 + cC dy^2) - cB dx dy
    float* P = params + i * PSTRIDE;
    P[0] = -0.5f * cA;                      // * x^2
    P[1] = -0.5f * cC;                      // * y^2
    P[2] = -cB;                             // * x*y
    P[3] = cA*px + cB*py;                   // * x
    P[4] = cC*py + cB*px;                   // * y
    P[5] = -0.5f*(cA*px*px + cC*py*py) - cB*px*py;  // * 1
    P[6] = 0.0f;
    P[7] = 0.0f;
    P[8]  = valid ? op : 0.0f;              // effective opacity
    P[9]  = cr; P[10] = cg; P[11] = cb;
    P[12] = 0.0f; P[13] = 0.0f; P[14] = 0.0f; P[15] = 0.0f;

    out_radii[i] = radii;
    out_depth[i] = tz;
}

// ---------------------------------------------------------------------------
// Kernel 2: single-block bitonic argsort by depth (ascending), then gather
// gaussian records into depth order (vectorized float4 gather).
// ---------------------------------------------------------------------------
__global__ __launch_bounds__(1024) void gs_sort_gather(
    const float* __restrict__ depth, const float* __restrict__ params,
    float* __restrict__ sorted)
{
    __shared__ float key[NPTS];
    __shared__ int   idx[NPTS];
    int tid = threadIdx.x;
    for (int i = tid; i < NPTS; i += 1024) { key[i] = depth[i]; idx[i] = i; }
    __syncthreads();

    for (int k = 2; k <= NPTS; k <<= 1) {
        for (int j = k >> 1; j > 0; j >>= 1) {
            for (int i = tid; i < NPTS; i += 1024) {
                int l = i ^ j;
                if (l > i) {
                    bool up = ((i & k) == 0);
                    float ki = key[i], kl = key[l];
                    if ((ki > kl) == up) {
                        key[i] = kl; key[l] = ki;
                        int t = idx[i]; idx[i] = idx[l]; idx[l] = t;
                    }
                }
            }
            __syncthreads();
        }
    }

    const float4* p4 = (const float4*)params;
    float4* s4 = (float4*)sorted;
    for (int i = tid; i < NPTS; i += 1024) {
        int src = idx[i];
#pragma unroll
        for (int f = 0; f < PSTRIDE / 4; ++f)
            s4[i * (PSTRIDE/4) + f] = p4[src * (PSTRIDE/4) + f];
    }
}

// ---------------------------------------------------------------------------
// Kernel 3: rasterize. One wave (32 lanes) per 32 pixels.
// Per 16-gaussian chunk: async global->LDS DMA of the next chunk's records
// (GLOBAL_LOAD_ASYNC_TO_LDS_B128, ASYNCcnt) overlaps 4x v_wmma_f32_16x16x4_f32
// computing the 16x32 power tile and the per-pixel ordered alpha blend.
// Dynamic LDS arena so async LDS byte offsets start at 0:
//   [0    .. 1023]  params buf 0 (256 f)
//   [1024 .. 2047]  params buf 1 (256 f)
//   [2048 .. 4095]  power tile   (512 f)
// ---------------------------------------------------------------------------
__device__ __forceinline__ float pix_feat(int pixel, int k) {
    float xx = (float)(pixel & (W_IMG - 1));
    float yy = (float)(pixel >> 7);
    switch (k) {
        case 0: return xx * xx;
        case 1: return yy * yy;
        case 2: return xx * yy;
        case 3: return xx;
        case 4: return yy;
        case 5: return 1.0f;
        default: return 0.0f;
    }
}

__global__ __launch_bounds__(32) void gs_rasterize(
    const float* __restrict__ sorted, float* __restrict__ out_rgb)
{
    extern __shared__ float smem[];             // 1024 floats (4 KB)

    const int lane    = threadIdx.x;            // 0..31
    const int pixBase = blockIdx.x * 32;
    const int pix     = pixBase + lane;
    const int halfSel = lane >> 4;              // 0: lanes 0-15, 1: lanes 16-31
    const int nn      = lane & 15;
    const int k0      = halfSel * 2;            // K-pair owned by this lane half

    // Loop-invariant B operands (pixel features), ISA 32-bit A/B layout:
    // lanes 0-15 hold K=0..1 in VGPR0..1, lanes 16-31 hold K=2..3.
    const int plo = pixBase + nn;
    const int phi = pixBase + 16 + nn;
    v2f bLo1 = { pix_feat(plo, k0),     pix_feat(plo, k0 + 1) };
    v2f bLo2 = { pix_feat(plo, 4 + k0), pix_feat(plo, 5 + k0) };
    v2f bHi1 = { pix_feat(phi, k0),     pix_feat(phi, k0 + 1) };
    v2f bHi2 = { pix_feat(phi, 4 + k0), pix_feat(phi, 5 + k0) };

    const unsigned laneB = (unsigned)lane * 16u;   // per-lane 16B row

    // Issue async DMA of chunk c into params buffer (c & 1).
    // INST_OFFSET is added to BOTH the LDS and global addresses, so one
    // vdst/vaddr pair + offset:512 covers this lane's two 16B rows.
    auto issue_chunk = [&](int c) {
        unsigned ldsOff = (unsigned)(c & 1) * (unsigned)CHUNK_BYTES + laneB;
        unsigned gOff   = (unsigned)c * (unsigned)CHUNK_BYTES + laneB;
        asm volatile(
            "global_load_async_to_lds_b128 %0, %1, %2\n\t"
            "global_load_async_to_lds_b128 %0, %1, %2 offset:512"
            :: "v"(ldsOff), "v"(gOff), "s"(sorted)
            : "memory");
    };

    issue_chunk(0);                             // ASYNCcnt = 2

    float* powb = smem + 2 * CHUNK * PSTRIDE;   // power tile region

    float T = 1.0f, accR = 0.0f, accG = 0.0f, accB = 0.0f;

    for (int chunk = 0; chunk < NCHUNK; ++chunk) {
        if (chunk + 1 < NCHUNK) {
            issue_chunk(chunk + 1);             // ASYNCcnt <= 4
            // async loads complete in order: <=2 outstanding means chunk's
            // own DMA has landed in LDS.
            asm volatile("s_wait_asynccnt 0x2" ::: "memory");
        } else {
            asm volatile("s_wait_asynccnt 0x0" ::: "memory");
        }

        const float* pbuf = smem + (chunk & 1) * (CHUNK * PSTRIDE);

        // A operand: 16 gaussians x 4 coeffs (this lane's K pair)
        v2f a1 = { pbuf[nn*PSTRIDE + k0],     pbuf[nn*PSTRIDE + k0 + 1] };
        v2f a2 = { pbuf[nn*PSTRIDE + 4 + k0], pbuf[nn*PSTRIDE + 5 + k0] };

        v8f dLo = {};
        dLo = __builtin_amdgcn_wmma_f32_16x16x4_f32(false, a1, false, bLo1,
                                                    (short)0, dLo, false, false);
        dLo = __builtin_amdgcn_wmma_f32_16x16x4_f32(false, a2, false, bLo2,
                                                    (short)0, dLo, false, false);
        v8f dHi = {};
        dHi = __builtin_amdgcn_wmma_f32_16x16x4_f32(false, a1, false, bHi1,
                                                    (short)0, dHi, false, false);
        dHi = __builtin_amdgcn_wmma_f32_16x16x4_f32(false, a2, false, bHi2,
                                                    (short)0, dHi, false, false);

        // D layout: VGPR v, lanes 0-15 -> (g=v, p=lane); lanes 16-31 -> (g=v+8)
#pragma unroll
        for (int v = 0; v < 8; ++v) {
            int g = v + halfSel * 8;
            powb[g * 32 + nn]      = dLo[v];
            powb[g * 32 + 16 + nn] = dHi[v];
        }
        // single wave per workgroup: LDS ops are in-order within the wave,
        // this only fences compiler scheduling of the store->load pair.
        __builtin_amdgcn_wave_barrier();

        // ordered alpha blend over the 16 gaussians of this chunk
#pragma unroll 4
        for (int gi = 0; gi < CHUNK; ++gi) {
            float power = powb[gi * 32 + lane];
            float4 oc = *(const float4*)(pbuf + gi * PSTRIDE + 8); // op,r,g,b
            float alpha = fminf(oc.x * __expf(fminf(power, 0.0f)), 0.99f);
            // keep-test written so NaN power/alpha -> alpha=0 (matches ref skip)
            bool keep = (power <= 0.0f) && (alpha >= (1.0f / 255.0f));
            alpha = keep ? alpha : 0.0f;
            float w = alpha * T;
            accR += w * oc.y; accG += w * oc.z; accB += w * oc.w;
            T *= (1.0f - alpha);
        }
        __builtin_amdgcn_wave_barrier();
    }

    out_rgb[0 * NPIX + pix] = accR;
    out_rgb[1 * NPIX + pix] = accG;
    out_rgb[2 * NPIX + pix] = accB;
}

// ---------------------------------------------------------------------------
extern "C" void kernel_launch(void* const* d_in, const int* in_sizes, int n_in,
                              void* d_out, int out_size, void* d_ws, size_t ws_size,
                              hipStream_t stream) {
    const float* pts    = (const float*)d_in[0];
    const float* scales = (const float*)d_in[1];
    const float* colors = (const float*)d_in[2];
    const float* opac   = (const float*)d_in[3];
    const float* rots   = (const float*)d_in[4];
    const float* view   = (const float*)d_in[5];
    const float* proj   = (const float*)d_in[6];
    (void)in_sizes; (void)n_in; (void)out_size; (void)ws_size;

    float* out       = (float*)d_out;
    float* out_rgb   = out;                        // 3*H*W
    float* out_radii = out + 3 * NPIX;             // N
    float* out_depth = out + 3 * NPIX + NPTS;      // N

    float* params = (float*)d_ws;                  // N * PSTRIDE
    float* sorted = params + NPTS * PSTRIDE;       // N * PSTRIDE

    gs_preprocess<<<(NPTS + 255) / 256, 256, 0, stream>>>(
        pts, scales, colors, opac, rots, view, proj,
        params, out_radii, out_depth);

    gs_sort_gather<<<1, 1024, 0, stream>>>(out_depth, params, sorted);

    // dynamic LDS: 2 x 1KB param buffers + 2KB power tile
    gs_rasterize<<<NPIX / 32, 32, 4096, stream>>>(sorted, out_rgb);
}